// DeepGMRCustom_1795296330061
// MI455X (gfx1250) — compile-verified
//
#include <hip/hip_runtime.h>
#include <hip/hip_bf16.h>
#include <math.h>

// Problem constants (from reference): B=8, D=3, N=65536, K=64
#define BQ 8
#define NQ 65536
#define KQ 64
#define EPSQ 1e-8f
#define CHUNKS 16
#define CHUNK (NQ / CHUNKS)   // 4096 points per block
#define LOG2E 1.4426950408889634f

typedef __attribute__((ext_vector_type(2))) float v2f;
typedef __attribute__((ext_vector_type(8))) float v8f;

// Raw v_exp_f32 (2^x), no denormal-range guard: exact fit for softmax where
// args <= 0 and sub-2^-126 contributions are negligible (flush-to-zero is fine).
__device__ __forceinline__ float fast_exp2(float x) {
    return __builtin_amdgcn_exp2f(x);
}

// ---------------------------------------------------------------------------
// Stage 1: per (batch, set, chunk) softmax-over-N statistics per cluster.
// logits tile (16 points x 64 clusters) via V_WMMA_F32_16X16X4_F32:
//   A = 16x4 f32  (points x [x,y,z,0])   lanes 0-15: K=0,1 ; lanes 16-31: K=2,3
//   B = 4x16 f32  ([x,y,z,0] x clusters) VGPR v: lanes 0-15 K=v, lanes 16-31 K=v+2
//   C/D 16x16 f32: VGPR v, lanes 0-15 -> M=v, N=lane ; lanes 16-31 -> M=v+8
// W pre-scaled by log2(e) -> softmax in v_exp_f32's native base-2 domain.
// Tile-group max hoisted -> one accumulator rescale per 8 logits.
// ---------------------------------------------------------------------------
__global__ __launch_bounds__(256) void gmm_stats_kernel(
    const float* __restrict__ Tf, const float* __restrict__ Sf,
    const float* __restrict__ Wk, float* __restrict__ part)
{
    const int lane  = threadIdx.x & 31;
    const int wave  = threadIdx.x >> 5;          // 0..7
    const int chunk = blockIdx.x & (CHUNKS - 1);
    const int set   = (blockIdx.x >> 4) & 1;     // 0 = template, 1 = source
    const int b     = blockIdx.x >> 5;
    const float* __restrict__ F =
        (set == 0 ? Tf : Sf) + (size_t)b * 3 * NQ;

    const int  col = lane & 15;
    const bool hi  = lane >= 16;

    // B fragments: W is (D=3, K=64) row-major; pad D to 4; fold in log2(e).
    v2f bf[4];
    #pragma unroll
    for (int g = 0; g < 4; ++g) {
        int k = g * 16 + col;
        if (!hi) { bf[g].x = LOG2E * Wk[0 * KQ + k]; bf[g].y = LOG2E * Wk[1 * KQ + k]; }
        else     { bf[g].x = LOG2E * Wk[2 * KQ + k]; bf[g].y = 0.0f; }
    }

    // base-2 online softmax state per 16-cluster group
    float m[4], Z[4], Sx[4], Sy[4], Sz[4];
    #pragma unroll
    for (int g = 0; g < 4; ++g) { m[g] = -3.0e38f; Z[g] = 0.f; Sx[g] = 0.f; Sy[g] = 0.f; Sz[g] = 0.f; }

    const int p_begin = chunk * CHUNK;
    const int ntiles  = CHUNK / 16;              // 256 tiles, 8 waves -> 32 each

    for (int t = wave; t < ntiles; t += 8) {
        const int p0 = p_begin + t * 16;
        // prefetch next tile's x-plane (global_prefetch_b8 on gfx1250)
        __builtin_prefetch(&F[p0 + 16 * 8], 0, 1);

        // A fragment (16 points x 4)
        v2f a;
        if (!hi) { a.x = F[p0 + col];          a.y = F[NQ + p0 + col]; }
        else     { a.x = F[2 * NQ + p0 + col]; a.y = 0.0f; }

        // coordinates of the 8 M-rows this lane owns in the C layout
        const int pb = p0 + (hi ? 8 : 0);
        float xc[8], yc[8], zc[8];
        #pragma unroll
        for (int i = 0; i < 8; ++i) {
            xc[i] = F[pb + i];
            yc[i] = F[NQ + pb + i];
            zc[i] = F[2 * NQ + pb + i];
        }

        const v8f cz = {0.f, 0.f, 0.f, 0.f, 0.f, 0.f, 0.f, 0.f};
        #pragma unroll
        for (int g = 0; g < 4; ++g) {
            v8f c = __builtin_amdgcn_wmma_f32_16x16x4_f32(
                false, a, false, bf[g], (short)0, cz, false, false);

            // hoisted tile-group max -> one accumulator rescale per 8 logits
            float tmax = c[0];
            #pragma unroll
            for (int i = 1; i < 8; ++i) tmax = fmaxf(tmax, c[i]);
            const float mn = fmaxf(m[g], tmax);
            const float sc = fast_exp2(m[g] - mn);
            m[g]  = mn;
            Z[g]  *= sc; Sx[g] *= sc; Sy[g] *= sc; Sz[g] *= sc;

            #pragma unroll
            for (int i = 0; i < 8; ++i) {
                const float e = fast_exp2(c[i] - mn);
                Z[g]  += e;
                Sx[g] = fmaf(e, xc[i], Sx[g]);
                Sy[g] = fmaf(e, yc[i], Sy[g]);
                Sz[g] = fmaf(e, zc[i], Sz[g]);
            }
        }
    }

    // merge lane L with lane L^16 (both hold the same cluster)
    #pragma unroll
    for (int g = 0; g < 4; ++g) {
        float m2 = __shfl_xor(m[g], 16, 32);
        float Z2 = __shfl_xor(Z[g], 16, 32);
        float X2 = __shfl_xor(Sx[g], 16, 32);
        float Y2 = __shfl_xor(Sy[g], 16, 32);
        float W2 = __shfl_xor(Sz[g], 16, 32);
        float mn = fmaxf(m[g], m2);
        float s1 = fast_exp2(m[g] - mn);
        float s2 = fast_exp2(m2 - mn);
        m[g]  = mn;
        Z[g]  = Z[g]  * s1 + Z2 * s2;
        Sx[g] = Sx[g] * s1 + X2 * s2;
        Sy[g] = Sy[g] * s1 + Y2 * s2;
        Sz[g] = Sz[g] * s1 + W2 * s2;
    }

    // cross-wave reduction in LDS
    __shared__ float red[8][64][5];
    if (!hi) {
        #pragma unroll
        for (int g = 0; g < 4; ++g) {
            int k = g * 16 + col;
            red[wave][k][0] = m[g];  red[wave][k][1] = Z[g];
            red[wave][k][2] = Sx[g]; red[wave][k][3] = Sy[g];
            red[wave][k][4] = Sz[g];
        }
    }
    __syncthreads();

    if (threadIdx.x < 64) {
        const int k = threadIdx.x;
        float mm = red[0][k][0], ZZ = red[0][k][1];
        float XX = red[0][k][2], YY = red[0][k][3], WW = red[0][k][4];
        for (int w = 1; w < 8; ++w) {
            float m2 = red[w][k][0], Z2 = red[w][k][1];
            float X2 = red[w][k][2], Y2 = red[w][k][3], W2 = red[w][k][4];
            float mn = fmaxf(mm, m2);
            float s1 = fast_exp2(mm - mn), s2 = fast_exp2(m2 - mn);
            mm = mn;
            ZZ = ZZ * s1 + Z2 * s2; XX = XX * s1 + X2 * s2;
            YY = YY * s1 + Y2 * s2; WW = WW * s1 + W2 * s2;
        }
        float* o = part + ((((size_t)b * 2 + set) * CHUNKS + chunk) * KQ + k) * 5;
        o[0] = mm; o[1] = ZZ; o[2] = XX; o[3] = YY; o[4] = WW;
    }
}

// ---------------------------------------------------------------------------
// Stage 2: merge chunk partials -> mu[b,set,k,3]   (base-2 log-sum-exp merge)
// softmax sums to 1 exactly => Npi = 1+eps, g = gamma/(1+2eps)
// ---------------------------------------------------------------------------
__global__ void reduce_mu_kernel(const float* __restrict__ part,
                                 float* __restrict__ mu)
{
    const int bs = blockIdx.x;     // b*2+set, 16 blocks
    const int k  = threadIdx.x;    // 64 threads
    float mm = -3.0e38f, ZZ = 0.f, XX = 0.f, YY = 0.f, WW = 0.f;
    for (int c = 0; c < CHUNKS; ++c) {
        const float* q = part + (((size_t)bs * CHUNKS + c) * KQ + k) * 5;
        float m2 = q[0], Z2 = q[1], X2 = q[2], Y2 = q[3], W2 = q[4];
        float mn = fmaxf(mm, m2);
        float s1 = fast_exp2(mm - mn), s2 = fast_exp2(m2 - mn);
        mm = mn;
        ZZ = ZZ * s1 + Z2 * s2; XX = XX * s1 + X2 * s2;
        YY = YY * s1 + Y2 * s2; WW = WW * s1 + W2 * s2;
    }
    const float inv = 1.0f / (ZZ * (1.0f + 2.0f * EPSQ));
    float* o = mu + ((size_t)bs * KQ + k) * 3;
    o[0] = XX * inv; o[1] = YY * inv; o[2] = WW * inv;
}

// ---------------------------------------------------------------------------
// Stage 3: per-batch rigid transform. pi is uniform = (1+eps)/N.
// R0 = orthogonal polar factor of Wm (== U@Vh of the SVD); reference's det
// fix scales R0's third column by sign(det R0).
// ---------------------------------------------------------------------------
__global__ void transform_kernel(const float* __restrict__ mu,
                                 float* __restrict__ RT)
{
    const int b = threadIdx.x;
    if (b >= BQ) return;
    const float* muT = mu + (size_t)(b * 2 + 0) * KQ * 3;  // template = "y"
    const float* muS = mu + (size_t)(b * 2 + 1) * KQ * 3;  // source   = "x"
    const float pi = (1.0f + EPSQ) / (float)NQ;

    float cx[3] = {0, 0, 0}, cy[3] = {0, 0, 0};
    for (int k = 0; k < KQ; ++k)
        for (int d = 0; d < 3; ++d) {
            cx[d] += muS[k * 3 + d];
            cy[d] += muT[k * 3 + d];
        }
    for (int d = 0; d < 3; ++d) { cx[d] *= pi; cy[d] *= pi; }

    float Wm[3][3] = {{0.f}};
    for (int k = 0; k < KQ; ++k) {
        float yv[3], xv[3];
        for (int d = 0; d < 3; ++d) {
            yv[d] = muT[k * 3 + d] - cy[d];
            xv[d] = muS[k * 3 + d] - cx[d];
        }
        for (int d = 0; d < 3; ++d)
            for (int e = 0; e < 3; ++e)
                Wm[d][e] = fmaf(pi * yv[d], xv[e], Wm[d][e]);
    }

    // Newton polar iteration X <- 0.5*(X + X^{-T}), X^{-T} = cof(X)/det(X)
    float fn = 0.f;
    for (int d = 0; d < 3; ++d)
        for (int e = 0; e < 3; ++e) fn += Wm[d][e] * Wm[d][e];
    fn = sqrtf(fn) + 1e-30f;
    float X[3][3];
    for (int d = 0; d < 3; ++d)
        for (int e = 0; e < 3; ++e) X[d][e] = Wm[d][e] / fn;

    float det = 0.f;
    for (int it = 0; it < 30; ++it) {
        float c00 = X[1][1]*X[2][2] - X[1][2]*X[2][1];
        float c01 = X[1][2]*X[2][0] - X[1][0]*X[2][2];
        float c02 = X[1][0]*X[2][1] - X[1][1]*X[2][0];
        float c10 = X[0][2]*X[2][1] - X[0][1]*X[2][2];
        float c11 = X[0][0]*X[2][2] - X[0][2]*X[2][0];
        float c12 = X[0][1]*X[2][0] - X[0][0]*X[2][1];
        float c20 = X[0][1]*X[1][2] - X[0][2]*X[1][1];
        float c21 = X[0][2]*X[1][0] - X[0][0]*X[1][2];
        float c22 = X[0][0]*X[1][1] - X[0][1]*X[1][0];
        det = X[0][0]*c00 + X[0][1]*c01 + X[0][2]*c02;
        float id = 0.5f / det;
        X[0][0] = 0.5f*X[0][0] + c00*id; X[0][1] = 0.5f*X[0][1] + c01*id; X[0][2] = 0.5f*X[0][2] + c02*id;
        X[1][0] = 0.5f*X[1][0] + c10*id; X[1][1] = 0.5f*X[1][1] + c11*id; X[1][2] = 0.5f*X[1][2] + c12*id;
        X[2][0] = 0.5f*X[2][0] + c20*id; X[2][1] = 0.5f*X[2][1] + c21*id; X[2][2] = 0.5f*X[2][2] + c22*id;
    }
    const float s = (det >= 0.f) ? 1.0f : -1.0f;   // sign(det(R0))
    float R[3][3];
    for (int d = 0; d < 3; ++d) {
        R[d][0] = X[d][0]; R[d][1] = X[d][1]; R[d][2] = s * X[d][2];
    }
    float* o = RT + (size_t)b * 12;
    for (int d = 0; d < 3; ++d) {
        o[d * 3 + 0] = R[d][0]; o[d * 3 + 1] = R[d][1]; o[d * 3 + 2] = R[d][2];
        o[9 + d] = cy[d] - (R[d][0]*cx[0] + R[d][1]*cx[1] + R[d][2]*cx[2]);
    }
}

// ---------------------------------------------------------------------------
// Stage 4: aligned[b,n,:] = R @ src[b,:,n] + T   (output layout B,N,3)
// ---------------------------------------------------------------------------
__global__ __launch_bounds__(256) void apply_kernel(
    const float* __restrict__ Sf, const float* __restrict__ RT,
    float* __restrict__ out)
{
    const size_t gid = (size_t)blockIdx.x * 256 + threadIdx.x;
    const int b = (int)(gid >> 16);          // N = 65536
    const int n = (int)(gid & (NQ - 1));
    const float* rt = RT + (size_t)b * 12;
    const float* F  = Sf + (size_t)b * 3 * NQ;
    const float x = F[n], y = F[NQ + n], z = F[2 * NQ + n];
    float* o = out + gid * 3;
    o[0] = fmaf(rt[0], x, fmaf(rt[1], y, fmaf(rt[2], z, rt[9])));
    o[1] = fmaf(rt[3], x, fmaf(rt[4], y, fmaf(rt[5], z, rt[10])));
    o[2] = fmaf(rt[6], x, fmaf(rt[7], y, fmaf(rt[8], z, rt[11])));
}

// ---------------------------------------------------------------------------
extern "C" void kernel_launch(void* const* d_in, const int* in_sizes, int n_in,
                              void* d_out, int out_size, void* d_ws, size_t ws_size,
                              hipStream_t stream)
{
    (void)in_sizes; (void)n_in; (void)out_size; (void)ws_size;
    const float* Tf = (const float*)d_in[0];   // (B,3,N)
    const float* Sf = (const float*)d_in[1];   // (B,3,N)
    const float* Wk = (const float*)d_in[2];   // (3,64)
    float* out = (float*)d_out;                // (B,N,3)

    float* part = (float*)d_ws;                              // 8*2*16*64*5 = 81920 f
    float* mu   = part + (size_t)BQ * 2 * CHUNKS * KQ * 5;   // 8*2*64*3   = 3072 f
    float* RT   = mu + (size_t)BQ * 2 * KQ * 3;              // 8*12       = 96 f

    gmm_stats_kernel<<<BQ * 2 * CHUNKS, 256, 0, stream>>>(Tf, Sf, Wk, part);
    reduce_mu_kernel<<<BQ * 2, KQ, 0, stream>>>(part, mu);
    transform_kernel<<<1, 32, 0, stream>>>(mu, RT);
    apply_kernel<<<(BQ * NQ) / 256, 256, 0, stream>>>(Sf, RT, out);
}